// IdealScoreMachine_26336739459425
// MI455X (gfx1250) — compile-verified
//
#include <hip/hip_runtime.h>
#include <math.h>

// CDNA5 / gfx1250: wave32, WMMA f32 16x16x4, async global->LDS DMA.
typedef __attribute__((ext_vector_type(2))) float v2f;
typedef __attribute__((ext_vector_type(8))) float v8f;
typedef int v4i_vs __attribute__((vector_size(4 * sizeof(int))));  // builtin's b128 type

#define AS1 __attribute__((address_space(1)))
#define AS3 __attribute__((address_space(3)))

#define DIMS     3072      // 3*32*32
#define BQ       16        // batch of x
#define NT       256       // images per workgroup tile
#define KC       16        // K-slice staged in LDS per step
#define NSLICES  (DIMS / KC)   // 192

#if defined(__HIP_DEVICE_COMPILE__) && __has_builtin(__builtin_amdgcn_global_load_async_to_lds_b128)
#define HAS_ASYNC_LDS 1
#else
#define HAS_ASYNC_LDS 0
#endif

static __device__ __forceinline__ void wait_asynccnt_le4() {
#if __has_builtin(__builtin_amdgcn_s_wait_asynccnt)
  __builtin_amdgcn_s_wait_asynccnt(4);
#else
  asm volatile("s_wait_asynccnt 0x4" ::: "memory");
#endif
}
static __device__ __forceinline__ void wait_asynccnt_0() {
#if __has_builtin(__builtin_amdgcn_s_wait_asynccnt)
  __builtin_amdgcn_s_wait_asynccnt(0);
#else
  asm volatile("s_wait_asynccnt 0x0" ::: "memory");
#endif
}

// ---------------------------------------------------------------------------
// Kernel 1: beta-schedule cumprod -> mu, sigma^2 ; x2[b] = ||x_b||^2
// ---------------------------------------------------------------------------
__global__ __launch_bounds__(512) void k_setup(const float* __restrict__ x,
                                               const int* __restrict__ t_ptr,
                                               float* __restrict__ hdr) {
  if (threadIdx.x == 0) {
    int t = t_ptr[0];
    float ab = 1.0f;
    for (int i = 0; i <= t; ++i) {
      float beta = 1e-4f + (0.02f - 1e-4f) * ((float)i / 999.0f);
      ab *= (1.0f - beta);
    }
    hdr[0] = sqrtf(ab);      // mu
    hdr[1] = 1.0f - ab;      // sigma^2
  }
  int b = threadIdx.x >> 5;
  int lane = threadIdx.x & 31;
  float s = 0.0f;
  for (int d = lane; d < DIMS; d += 32) {
    float v = x[b * DIMS + d];
    s += v * v;
  }
  for (int off = 16; off > 0; off >>= 1) s += __shfl_down(s, off, 32);
  if (lane == 0) hdr[2 + b] = s;
}

// ---------------------------------------------------------------------------
// Kernel 2: fused  G = x*tile^T  ->  w = exp(...)  ->  acc += w*tile
// One workgroup = one 256-image tile, 256 threads = 8 wave32.
// ---------------------------------------------------------------------------
__global__ __launch_bounds__(256) void k_main(const float* __restrict__ x,
                                              const float* __restrict__ images,
                                              const float* __restrict__ hdr,
                                              float* __restrict__ sumw,
                                              float* __restrict__ acc,
                                              int N) {
  __shared__ float s_tile[2][NT * KC];  // 32 KB, double-buffered K-slices
  __shared__ float s_w[BQ * NT];        // 16 KB, G-tile then w-tile in place
  __shared__ float s_red[256];          // 1 KB, sum_w reduction

  const int tid  = threadIdx.x;
  const int wave = tid >> 5;
  const int lane = tid & 31;
  const int lid  = lane & 15;   // fragment row/col index
  const int half = lane >> 4;   // lane-half selects K pair / M half
  const long tile0 = (long)blockIdx.x * NT;
  const bool full = (tile0 + NT) <= (long)N;

  const float mu     = hdr[0];
  const float sigma2 = hdr[1];

  // ---- Phase 1: G[16 x 256] = x[16 x D] * tile[256 x D]^T  (WMMA f32) ----
  v8f c0 = {};
  v8f c1 = {};
  const int n0 = wave * 32;     // two N-blocks of 16 images per wave

  // Slice mapping: 256 threads cover 256 rows x 16 floats; lanes 0..3 of a
  // row-group handle consecutive float4 (64B contiguous per row).
  const int lrow = tid >> 2;
  const int lcol = (tid & 3) * 4;
  float i2acc = 0.0f;           // thread tid accumulates ||image[tile0+tid]||^2

  // Tail blocks: rows >= N are never DMA'd -> persist zeros in both buffers.
  if (!full) {
#pragma unroll
    for (int g = 0; g < 4; ++g) {
      long row = tile0 + lrow + g * 64;
      if (row >= (long)N) {
        float4 z = make_float4(0.f, 0.f, 0.f, 0.f);
        *(float4*)(&s_tile[0][(lrow + g * 64) * KC + lcol]) = z;
        *(float4*)(&s_tile[1][(lrow + g * 64) * KC + lcol]) = z;
      }
    }
  }
  __syncthreads();

  // WMMA over one staged K-slice (A: 16x4 of x via L2, B: 4x16 of tile in LDS)
  auto compute_slice = [&](int buf, int ks) {
#pragma unroll
    for (int j = 0; j < KC; j += 4) {
      float4 v = *(const float4*)(&s_tile[buf][tid * KC + j]);
      i2acc += v.x * v.x + v.y * v.y + v.z * v.z + v.w * v.w;
    }
#pragma unroll
    for (int k = 0; k < KC; k += 4) {
      v2f a  = *(const v2f*)(x + lid * DIMS + ks + k + 2 * half);
      v2f b0 = *(const v2f*)(&s_tile[buf][(n0 + lid) * KC + k + 2 * half]);
      v2f b1 = *(const v2f*)(&s_tile[buf][(n0 + 16 + lid) * KC + k + 2 * half]);
      c0 = __builtin_amdgcn_wmma_f32_16x16x4_f32(false, a, false, b0,
                                                 (short)0, c0, false, false);
      c1 = __builtin_amdgcn_wmma_f32_16x16x4_f32(false, a, false, b1,
                                                 (short)0, c1, false, false);
    }
  };

#if HAS_ASYNC_LDS
  // Async DMA path: GLOBAL_LOAD_ASYNC_TO_LDS_B128 (ASYNCcnt), no VGPR bounce.
  auto issue_slice = [&](int ks, int buf) {
#pragma unroll
    for (int g = 0; g < 4; ++g) {
      long row = tile0 + lrow + g * 64;
      if (row < (long)N) {
        __builtin_amdgcn_global_load_async_to_lds_b128(
            (AS1 v4i_vs*)(images + row * (long)DIMS + ks + lcol),
            (AS3 v4i_vs*)(&s_tile[buf][(lrow + g * 64) * KC + lcol]),
            0, 0);
      }
    }
  };

  issue_slice(0, 0);
  for (int s = 0; s < NSLICES; ++s) {
    const int buf = s & 1;
    const int ks = s * KC;
    if (s + 1 < NSLICES) {
      issue_slice(ks + KC, (s + 1) & 1);
      wait_asynccnt_le4();      // this wave's slice-s DMAs have landed
    } else {
      wait_asynccnt_0();
    }
    __syncthreads();            // every wave's slice-s data visible
    compute_slice(buf, ks);
    __syncthreads();            // all reads of buf done before re-targeting it
  }
#else
  // Fallback: global_load_b128 -> VGPR -> ds_store_b128, double buffered.
  float4 pref[4];
  auto load_slice = [&](int ks) {
#pragma unroll
    for (int g = 0; g < 4; ++g) {
      long row = tile0 + lrow + g * 64;
      if (row < (long)N) {
        pref[g] = *(const float4*)(images + row * (long)DIMS + ks + lcol);
      } else {
        pref[g] = make_float4(0.f, 0.f, 0.f, 0.f);
      }
    }
  };
  auto store_slice = [&](int buf) {
#pragma unroll
    for (int g = 0; g < 4; ++g) {
      *(float4*)(&s_tile[buf][(lrow + g * 64) * KC + lcol]) = pref[g];
    }
  };

  load_slice(0);
  store_slice(0);
  __syncthreads();
  for (int s = 0; s < NSLICES; ++s) {
    const int buf = s & 1;
    const int ks = s * KC;
    if (s + 1 < NSLICES) load_slice(ks + KC);
    compute_slice(buf, ks);
    if (s + 1 < NSLICES) store_slice((s + 1) & 1);
    __syncthreads();
  }
#endif

  // Spill G fragments to LDS. C/D layout: M = v + 8*half, N = lid.
#pragma unroll
  for (int v = 0; v < 8; ++v) {
    s_w[(v + 8 * half) * NT + n0 + lid]      = c0[v];
    s_w[(v + 8 * half) * NT + n0 + 16 + lid] = c1[v];
  }
  __syncthreads();

  // ---- Phase 2: w = exp((2 mu G - x2 - mu^2 i2) / (2 sigma^2)) ----
  {
    const float inv2s2 = 0.5f / sigma2;
    const float mu2 = mu * mu;
    const bool valid = (tile0 + tid) < N;
#pragma unroll 4
    for (int b = 0; b < BQ; ++b) {
      float g   = s_w[b * NT + tid];
      float x2b = hdr[2 + b];
      float arg = (2.0f * mu * g - x2b - mu2 * i2acc) * inv2s2;
      s_w[b * NT + tid] = valid ? __expf(arg) : 0.0f;
    }
  }
  __syncthreads();

  // sum_w: 16 threads per batch row, then one global atomic per row
  {
    int b = tid >> 4, t2 = tid & 15;
    float p = 0.0f;
    for (int n = t2; n < NT; n += 16) p += s_w[b * NT + n];
    s_red[tid] = p;
  }
  __syncthreads();
  if (tid < BQ) {
    float ssum = 0.0f;
#pragma unroll
    for (int j = 0; j < 16; ++j) ssum += s_red[tid * 16 + j];
    atomicAdd(&sumw[tid], ssum);
  }

  // ---- Phase 3: acc[16 x D] += w[16 x 256] * tile[256 x D]  (WMMA f32) ----
  // Tile was just streamed -> B-fragment reads hit L2. A = w from LDS.
  if (full) {
    for (int db = wave; db < DIMS / 16; db += 8) {
      const int d0 = db * 16;
      v8f cacc = {};
      const float* bp = images + (tile0 + 2 * half) * (long)DIMS + d0 + lid;
#pragma unroll 8
      for (int k0 = 0; k0 < NT; k0 += 4) {
        v2f a = *(const v2f*)(&s_w[lid * NT + k0 + 2 * half]);
        v2f b;
        b.x = bp[(long)k0 * DIMS];
        b.y = bp[(long)(k0 + 1) * DIMS];
        cacc = __builtin_amdgcn_wmma_f32_16x16x4_f32(false, a, false, b,
                                                     (short)0, cacc, false, false);
      }
#pragma unroll
      for (int v = 0; v < 8; ++v) {
        atomicAdd(&acc[(v + 8 * half) * DIMS + d0 + lid], cacc[v]);
      }
    }
  } else {
    for (int db = wave; db < DIMS / 16; db += 8) {
      const int d0 = db * 16;
      v8f cacc = {};
#pragma unroll 4
      for (int k0 = 0; k0 < NT; k0 += 4) {
        v2f a = *(const v2f*)(&s_w[lid * NT + k0 + 2 * half]);
        long r0 = tile0 + k0 + 2 * half;
        v2f b;
        b.x = (r0 < N)     ? images[r0 * (long)DIMS + d0 + lid]       : 0.0f;
        b.y = (r0 + 1 < N) ? images[(r0 + 1) * (long)DIMS + d0 + lid] : 0.0f;
        cacc = __builtin_amdgcn_wmma_f32_16x16x4_f32(false, a, false, b,
                                                     (short)0, cacc, false, false);
      }
#pragma unroll
      for (int v = 0; v < 8; ++v) {
        atomicAdd(&acc[(v + 8 * half) * DIMS + d0 + lid], cacc[v]);
      }
    }
  }
}

// ---------------------------------------------------------------------------
// Kernel 3: out = -(sum_w * x - mu * acc) / (sigma^2 * sum_w + 1e-8)
// ---------------------------------------------------------------------------
__global__ __launch_bounds__(256) void k_final(const float* __restrict__ x,
                                               const float* __restrict__ hdr,
                                               const float* __restrict__ sumw,
                                               const float* __restrict__ acc,
                                               float* __restrict__ out) {
  int i = blockIdx.x * 256 + threadIdx.x;
  int b = i / DIMS;
  float mu = hdr[0], sigma2 = hdr[1];
  float sw = sumw[b];
  out[i] = -(sw * x[i] - mu * acc[i]) / (sigma2 * sw + 1e-8f);
}

// ---------------------------------------------------------------------------
extern "C" void kernel_launch(void* const* d_in, const int* in_sizes, int n_in,
                              void* d_out, int out_size, void* d_ws, size_t ws_size,
                              hipStream_t stream) {
  const float* x      = (const float*)d_in[0];
  const float* images = (const float*)d_in[1];
  const int*   t      = (const int*)d_in[2];
  const int N = in_sizes[1] / DIMS;  // 50000

  // workspace layout (floats): [0..31] header (mu, sigma2, x2[16]),
  // [32..47] sum_w, [64..64+16*3072) acc
  float* hdr  = (float*)d_ws;
  float* sumw = hdr + 32;
  float* acc  = hdr + 64;

  (void)hipMemsetAsync(sumw, 0, (size_t)(32 + BQ * DIMS) * sizeof(float), stream);
  k_setup<<<1, 512, 0, stream>>>(x, t, hdr);
  int ntiles = (N + NT - 1) / NT;
  k_main<<<ntiles, 256, 0, stream>>>(x, images, hdr, sumw, acc, N);
  k_final<<<(BQ * DIMS) / 256, 256, 0, stream>>>(x, hdr, sumw, acc, (float*)d_out);
}